// LDS_10617159155781
// MI455X (gfx1250) — compile-verified
//
#include <hip/hip_runtime.h>

// h_t = h_{t-1} @ A + x_t @ B ; y_t = h_t @ C   (fp32, chunked parallel scan)
// MI455X: f32 WMMA 16x16x4 tiles + async global->LDS staging on ASYNCcnt.
#define SEQ     4096
#define BATCH   8
#define DIM     256
#define CHUNK   16
#define NCHUNK  (SEQ / CHUNK)      // 256
#define MATS    (DIM * DIM)        // 65536
#define KSTEPS  (DIM / 4)          // 64 K-steps of the f32 16x16x4 WMMA
#define HSTRIDE 260                // LDS row stride (floats), avoids bank conflicts
#define XBUFROW (BATCH * HSTRIDE)  // one staged [8 x 256(+pad)] slab

typedef __attribute__((ext_vector_type(2))) float v2f;
typedef __attribute__((ext_vector_type(8))) float v8f;

// D = A(16x4) * B(4x16) + C, f32, wave32
static __device__ __forceinline__ v8f wmma4(v2f a, v2f b, v8f c) {
  return __builtin_amdgcn_wmma_f32_16x16x4_f32(false, a, false, b, (short)0, c,
                                               false, false);
}

// A-operand: row-major M x K matrix, stride DIM. lane holds (row=lo, k=2hi..2hi+1)
static __device__ __forceinline__ v2f ldA(const float* __restrict__ M, int row, int k) {
  v2f a; a.x = M[row * DIM + k]; a.y = M[row * DIM + k + 1]; return a;
}
// B-operand: row-major K x N matrix, stride DIM. lane holds (k=2hi..2hi+1, col=lo)
static __device__ __forceinline__ v2f ldB(const float* __restrict__ M, int k, int col) {
  v2f b; b.x = M[k * DIM + col]; b.y = M[(k + 1) * DIM + col]; return b;
}

// Async stage one [8 x 256] f32 slab (global, stride DIM) into LDS rows of
// HSTRIDE floats. 512 threads -> one b128 transfer each; tracked on ASYNCcnt.
static __device__ __forceinline__ void stage8x256_async(const float* __restrict__ g,
                                                        const float* lds_dst,
                                                        int tid) {
  int b   = tid >> 6;            // 0..7
  int seg = (tid & 63) << 2;     // float offset 0,4,...,252
  const float* src = g + b * DIM + seg;
  unsigned int l = (unsigned int)(unsigned long long)(lds_dst + b * HSTRIDE + seg);
  asm volatile("global_load_async_to_lds_b128 %0, %1, off"
               :: "v"(l), "v"(src) : "memory");
}
static __device__ __forceinline__ void wait_async0() {
  asm volatile("s_wait_asynccnt 0x0" ::: "memory");
}

// Generic 256x256x256 GEMM 16x16 tile (one wave): D[m0:,n0:] = Am @ Bm
static __device__ __forceinline__ void gemm256_tile(const float* __restrict__ Am,
                                                    const float* __restrict__ Bm,
                                                    float* __restrict__ Dm,
                                                    int m0, int n0, int lane) {
  int lo = lane & 15, hi = lane >> 4;
  v8f a0 = {}, a1 = {};
#pragma unroll 8
  for (int kk = 0; kk < KSTEPS; kk += 2) {
    int k = kk * 4 + 2 * hi;
    a0 = wmma4(ldA(Am, m0 + lo, k),     ldB(Bm, k,     n0 + lo), a0);
    a1 = wmma4(ldA(Am, m0 + lo, k + 4), ldB(Bm, k + 4, n0 + lo), a1);
  }
  v8f acc = a0 + a1;
#pragma unroll
  for (int r = 0; r < 8; ++r)
    Dm[(m0 + r + 8 * hi) * DIM + n0 + lo] = acc[r];
}

// --- kernel 1: pow[0] = I, pow[1] = A -------------------------------------
__global__ void k_init_pow(const float* __restrict__ A, float* __restrict__ powb) {
  int idx = blockIdx.x * blockDim.x + threadIdx.x;   // 65536 threads
  int r = idx >> 8, c = idx & 255;
  powb[idx]        = (r == c) ? 1.0f : 0.0f;
  powb[MATS + idx] = A[idx];
}

// --- kernel 2: log-depth powers: pow[m+j] = pow[j] @ pow[m], j=1..m -------
__global__ void k_pow_step(float* __restrict__ powb, int m) {
  int wave = (int)((blockIdx.x * blockDim.x + threadIdx.x) >> 5);
  int lane = threadIdx.x & 31;
  if (wave >= m * 256) return;                 // wave-uniform
  int j = (wave >> 8) + 1;                     // 1..m
  int t = wave & 255;
  gemm256_tile(powb + j * MATS, powb + m * MATS, powb + (m + j) * MATS,
               (t >> 4) * 16, (t & 15) * 16, lane);
}

// --- kernel 3: bap[j] = B @ A^j, j = 0..15 --------------------------------
__global__ void k_bapow(const float* __restrict__ Bmat, const float* __restrict__ powb,
                        float* __restrict__ bap) {
  int wave = (int)((blockIdx.x * blockDim.x + threadIdx.x) >> 5);  // 4096 waves
  int lane = threadIdx.x & 31;
  int j = wave >> 8;                            // 0..15
  int t = wave & 255;
  gemm256_tile(Bmat, powb + j * MATS, bap + j * MATS,
               (t >> 4) * 16, (t & 15) * 16, lane);
}

// --- kernel 4: V[c] = sum_j x[:,c*16+j,:] @ bap[15-j]  (chunk input agg) --
__global__ void k_chunk_agg(const float* __restrict__ x, const float* __restrict__ bap,
                            float* __restrict__ V) {
  int wave = (int)((blockIdx.x * blockDim.x + threadIdx.x) >> 5);  // 4096 waves
  int lane = threadIdx.x & 31;
  int c  = wave >> 4;                     // chunk 0..255
  int n0 = (wave & 15) * 16;
  int lo = lane & 15, hi = lane >> 4;
  v8f a0 = {}, a1 = {};
#pragma unroll 1
  for (int j = 0; j < CHUNK; ++j) {
    int s = c * CHUNK + j;
    const float* Bm = bap + (CHUNK - 1 - j) * MATS;
#pragma unroll 8
    for (int kk = 0; kk < KSTEPS; kk += 2) {
      int k = kk * 4 + 2 * hi;
      v2f a, a2;
      a.x  = (lo < 8) ? x[(lo * SEQ + s) * DIM + k] : 0.0f;
      a.y  = (lo < 8) ? x[(lo * SEQ + s) * DIM + k + 1] : 0.0f;
      a2.x = (lo < 8) ? x[(lo * SEQ + s) * DIM + k + 4] : 0.0f;
      a2.y = (lo < 8) ? x[(lo * SEQ + s) * DIM + k + 5] : 0.0f;
      a0 = wmma4(a,  ldB(Bm, k,     n0 + lo), a0);
      a1 = wmma4(a2, ldB(Bm, k + 4, n0 + lo), a1);
    }
  }
  v8f acc = a0 + a1;
  if (hi == 0) {
#pragma unroll
    for (int r = 0; r < 8; ++r)
      V[(c * BATCH + r) * DIM + n0 + lo] = acc[r];   // rows = batch 0..7
  }
}

// --- kernel 5: sequential boundary scan: G[c]=h ; h = h@A^16 + V[c] -------
// V[c+1] is prefetched async into an LDS double buffer so the L2 read is off
// the 256-step dependent critical path.
__global__ void __launch_bounds__(512) k_boundary(const float* __restrict__ powb,
                                                  const float* __restrict__ V,
                                                  const float* __restrict__ h0,
                                                  float* __restrict__ G) {
  __shared__ float hbuf[16 * HSTRIDE];
  __shared__ float vbuf[2 * XBUFROW];
  int tid = threadIdx.x;
  int lane = tid & 31, w = tid >> 5;
  int lo = lane & 15, hi = lane >> 4, n0 = w * 16;

  for (int idx = tid; idx < 16 * DIM; idx += 512) {
    int r = idx >> 8, col = idx & 255;
    hbuf[r * HSTRIDE + col] = (r < 8) ? h0[col] : 0.0f;   // rows 8..15 = zero pad
  }
  // Preload this wave's B-operand slab of A^16 into registers (64 x v2f)
  const float* A16 = powb + CHUNK * MATS;
  v2f bsl[KSTEPS];
#pragma unroll
  for (int kk = 0; kk < KSTEPS; ++kk) {
    int k = kk * 4 + 2 * hi;
    bsl[kk].x = A16[k * DIM + n0 + lo];
    bsl[kk].y = A16[(k + 1) * DIM + n0 + lo];
  }
  stage8x256_async(V, vbuf, tid);                 // prefetch V[0]
  __syncthreads();

#pragma unroll 1
  for (int c = 0; c < NCHUNK; ++c) {
    wait_async0();                                // vbuf[c&1] landed (this wave)
    __syncthreads();                              // ... and every other wave's part
    if (c + 1 < NCHUNK)                           // prefetch V[c+1]
      stage8x256_async(V + (c + 1) * (BATCH * DIM), vbuf + ((c + 1) & 1) * XBUFROW, tid);
    // emit boundary state G_c = h_{c*16-1}
    for (int idx = tid; idx < BATCH * DIM; idx += 512)
      G[c * (BATCH * DIM) + idx] = hbuf[(idx >> 8) * HSTRIDE + (idx & 255)];
    // h @ A^16 for this wave's 16-column slab; 4 interleaved accumulators
    v8f ac[4] = {};
#pragma unroll
    for (int kk = 0; kk < KSTEPS; ++kk) {
      int k = kk * 4 + 2 * hi;
      v2f a; a.x = hbuf[lo * HSTRIDE + k]; a.y = hbuf[lo * HSTRIDE + k + 1];
      ac[kk & 3] = wmma4(a, bsl[kk], ac[kk & 3]);
    }
    v8f acc = (ac[0] + ac[1]) + (ac[2] + ac[3]);
    const float* vb = vbuf + (c & 1) * XBUFROW;
#pragma unroll
    for (int r = 0; r < 8; ++r)
      acc[r] += (hi == 0) ? vb[r * HSTRIDE + n0 + lo] : 0.0f;
    __syncthreads();
#pragma unroll
    for (int r = 0; r < 8; ++r)
      hbuf[(r + 8 * hi) * HSTRIDE + n0 + lo] = acc[r];
    __syncthreads();
  }
}

// --- kernel 6: per-chunk interior replay + fused y = h @ C ----------------
// x_s rows are staged once per block via async b128 copies (double-buffered),
// instead of 16 waves each re-reading them from global on the critical path.
__global__ void __launch_bounds__(512) k_interior(const float* __restrict__ x,
                                                  const float* __restrict__ powb,
                                                  const float* __restrict__ Bmat,
                                                  const float* __restrict__ Cmat,
                                                  const float* __restrict__ G,
                                                  float* __restrict__ y) {
  __shared__ float hbuf[16 * HSTRIDE];
  __shared__ float xbuf[2 * XBUFROW];
  int c = blockIdx.x;
  int tid = threadIdx.x;
  int lane = tid & 31, w = tid >> 5;
  int lo = lane & 15, hi = lane >> 4, n0 = w * 16;

  for (int idx = tid; idx < 16 * DIM; idx += 512) {
    int r = idx >> 8, col = idx & 255;
    hbuf[r * HSTRIDE + col] = (r < 8) ? G[c * (BATCH * DIM) + idx] : 0.0f;
  }
  // Preload A slab (state matrix) into registers; B and C stream from L2.
  const float* Amat = powb + MATS;
  v2f asl[KSTEPS];
#pragma unroll
  for (int kk = 0; kk < KSTEPS; ++kk) {
    int k = kk * 4 + 2 * hi;
    asl[kk].x = Amat[k * DIM + n0 + lo];
    asl[kk].y = Amat[(k + 1) * DIM + n0 + lo];
  }
  // x rows are [b][s][k]: row b of step s lives at x + (b*SEQ + s)*DIM, but
  // stage8x256 wants stride DIM between rows -> pass base x + s*DIM with row
  // stride SEQ*DIM folded in by a custom mapping below.
  {  // prefetch x_{s0}
    int b = tid >> 6, seg = (tid & 63) << 2;
    const float* src = x + (b * SEQ + c * CHUNK) * DIM + seg;
    unsigned int l = (unsigned int)(unsigned long long)(xbuf + b * HSTRIDE + seg);
    asm volatile("global_load_async_to_lds_b128 %0, %1, off"
                 :: "v"(l), "v"(src) : "memory");
  }
  __syncthreads();

#pragma unroll 1
  for (int j = 0; j < CHUNK; ++j) {
    int s = c * CHUNK + j;
    wait_async0();
    __syncthreads();                              // xbuf[j&1] visible to all waves
    if (j + 1 < CHUNK) {                          // prefetch x_{s+1}
      int b = tid >> 6, seg = (tid & 63) << 2;
      const float* src = x + (b * SEQ + s + 1) * DIM + seg;
      unsigned int l =
          (unsigned int)(unsigned long long)(xbuf + ((j + 1) & 1) * XBUFROW + b * HSTRIDE + seg);
      asm volatile("global_load_async_to_lds_b128 %0, %1, off"
                   :: "v"(l), "v"(src) : "memory");
    }
    const float* xb = xbuf + (j & 1) * XBUFROW;
    v8f ac[4] = {};
    // x_s @ B (LDS A-operand, global B-operand)
#pragma unroll 8
    for (int kk = 0; kk < KSTEPS; ++kk) {
      int k = kk * 4 + 2 * hi;
      v2f a;
      a.x = (lo < 8) ? xb[lo * HSTRIDE + k] : 0.0f;
      a.y = (lo < 8) ? xb[lo * HSTRIDE + k + 1] : 0.0f;
      ac[kk & 3] = wmma4(a, ldB(Bmat, k, n0 + lo), ac[kk & 3]);
    }
    // + h_{s-1} @ A (LDS A-operand, register-slab B-operand)
#pragma unroll
    for (int kk = 0; kk < KSTEPS; ++kk) {
      int k = kk * 4 + 2 * hi;
      v2f a; a.x = hbuf[lo * HSTRIDE + k]; a.y = hbuf[lo * HSTRIDE + k + 1];
      ac[kk & 3] = wmma4(a, asl[kk], ac[kk & 3]);
    }
    v8f hn = (ac[0] + ac[1]) + (ac[2] + ac[3]);
    __syncthreads();
#pragma unroll
    for (int r = 0; r < 8; ++r)
      hbuf[(r + 8 * hi) * HSTRIDE + n0 + lo] = hn[r];
    __syncthreads();
    // y_s = h_s @ C
    v8f yc[4] = {};
#pragma unroll 8
    for (int kk = 0; kk < KSTEPS; ++kk) {
      int k = kk * 4 + 2 * hi;
      v2f a; a.x = hbuf[lo * HSTRIDE + k]; a.y = hbuf[lo * HSTRIDE + k + 1];
      yc[kk & 3] = wmma4(a, ldB(Cmat, k, n0 + lo), yc[kk & 3]);
    }
    v8f yv = (yc[0] + yc[1]) + (yc[2] + yc[3]);
    if (hi == 0) {
#pragma unroll
      for (int r = 0; r < 8; ++r)
        y[(r * SEQ + s) * DIM + n0 + lo] = yv[r];   // y[b][s][o]
    }
  }
}

extern "C" void kernel_launch(void* const* d_in, const int* in_sizes, int n_in,
                              void* d_out, int out_size, void* d_ws, size_t ws_size,
                              hipStream_t stream) {
  const float* x  = (const float*)d_in[0];
  const float* A  = (const float*)d_in[1];
  const float* B  = (const float*)d_in[2];
  const float* C  = (const float*)d_in[3];
  const float* h0 = (const float*)d_in[4];
  float* y = (float*)d_out;

  float* powb = (float*)d_ws;                     // 17 * 65536 floats
  float* bap  = powb + 17 * MATS;                 // 16 * 65536 floats
  float* V    = bap + 16 * MATS;                  // NCHUNK*8*256 floats
  float* G    = V + NCHUNK * BATCH * DIM;         // NCHUNK*8*256 floats

  k_init_pow<<<256, 256, 0, stream>>>(A, powb);
  k_pow_step<<<1 * 32, 256, 0, stream>>>(powb, 1);   // A^2
  k_pow_step<<<2 * 32, 256, 0, stream>>>(powb, 2);   // A^3..A^4
  k_pow_step<<<4 * 32, 256, 0, stream>>>(powb, 4);   // A^5..A^8
  k_pow_step<<<8 * 32, 256, 0, stream>>>(powb, 8);   // A^9..A^16
  k_bapow<<<512, 256, 0, stream>>>(B, powb, bap);
  k_chunk_agg<<<512, 256, 0, stream>>>(x, bap, V);
  k_boundary<<<1, 512, 0, stream>>>(powb, V, h0, G);
  k_interior<<<NCHUNK, 512, 0, stream>>>(x, powb, B, C, G, y);
}